// BoundaryKDV1_54812372631996
// MI455X (gfx1250) — compile-verified
//
#include <hip/hip_runtime.h>
#include <hip/hip_bf16.h>
#include <stdint.h>

#define NC   14
#define DIM  96
#define PP   (96*96*96)      /* 884736 voxels per batch */
#define NTOT (2*PP)          /* 1769472 */
#define G2   2048            /* fixed grid for main kernel -> partial rows */
#define BLK  256
#define PSTRIDE 64           /* floats per partial row: [num e=0..31 | cnt e=0..31], e = b*16+k */

typedef __attribute__((ext_vector_type(2))) float v2f;
typedef __attribute__((ext_vector_type(8))) float v8f;

// ---------------- Stage 1: compact int labels -> u8 (1.7MB, L2-resident) ----------------
__global__ void bkd_compact_labels(const int* __restrict__ gt, unsigned char* __restrict__ lab) {
    int i = blockIdx.x * blockDim.x + threadIdx.x;
    if (i < NTOT) lab[i] = (unsigned char)gt[i];
}

// ---------------- Stage 2: fused KL + boundary stencil + hierarchical reduce ----------------
__global__ __launch_bounds__(BLK) void bkd_main(
    const float* __restrict__ S, const float* __restrict__ T,
    const unsigned char* __restrict__ lab, float* __restrict__ partials)
{
    __shared__ float red[(BLK/32) * PSTRIDE];
    const int tid  = threadIdx.x;
    const int lane = tid & 31;
    const int wv   = tid >> 5;

    // per-lane accumulators: lane k (k<14) owns class k; b0/b1 per batch
    float rn0 = 0.f, rn1 = 0.f, rc0 = 0.f, rc1 = 0.f;

    for (int idx = blockIdx.x * BLK + tid; idx < NTOT; idx += G2 * BLK) {
        const int b = (idx >= PP) ? 1 : 0;          // wave-uniform (PP % 32 == 0)
        const int p = idx - b * PP;
        const int d = p % DIM;
        const int w = (p / DIM) % DIM;
        const int h = p / (DIM * DIM);

        // ---- boundary stencil: present-class bitmask over 26 in-bounds neighbors ----
        const unsigned char* lb = lab + (size_t)b * PP;
        unsigned present = 0u;
        #pragma unroll
        for (int dh = -1; dh <= 1; ++dh)
        #pragma unroll
        for (int dw = -1; dw <= 1; ++dw)
        #pragma unroll
        for (int dd = -1; dd <= 1; ++dd) {
            if (dh == 0 && dw == 0 && dd == 0) continue;
            if ((unsigned)(h + dh) < (unsigned)DIM &&
                (unsigned)(w + dw) < (unsigned)DIM &&
                (unsigned)(d + dd) < (unsigned)DIM) {
                present |= 1u << lb[p + (dh * DIM + dw) * DIM + dd];
            }
        }
        const bool interior = (h >= 1 && h <= DIM - 2 && w >= 1 && w <= DIM - 2 &&
                               d >= 1 && d <= DIM - 2);
        // conv==26 (excluded) iff interior AND single-class neighborhood
        const unsigned bmask =
            (interior && ((present & (present - 1u)) == 0u)) ? 0u : present;

        // ---- per-voxel KL over 14 classes (one streaming pass, registers) ----
        const float* sp = S + ((size_t)b * NC) * PP + p;
        const float* tp = T + ((size_t)b * NC) * PP + p;
        float sv[NC], tv[NC];
        #pragma unroll
        for (int c = 0; c < NC; ++c) {
            sv[c] = sp[(size_t)c * PP];
            tv[c] = tp[(size_t)c * PP];
        }
        float maxS = sv[0], maxT = tv[0];
        #pragma unroll
        for (int c = 1; c < NC; ++c) {
            maxS = fmaxf(maxS, sv[c]);
            maxT = fmaxf(maxT, tv[c]);
        }
        float sumS = 0.f, sumT = 0.f, dot = 0.f;
        #pragma unroll
        for (int c = 0; c < NC; ++c) {
            float et = expf(tv[c] - maxT);
            sumT += et;
            dot  += et * (tv[c] - sv[c]);
            sumS += expf(sv[c] - maxS);
        }
        const float kl = (maxS - maxT) + (logf(sumS) - logf(sumT)) + dot / sumT;

        // ---- per-class wave32 reduction (shuffles + ballot popcount) ----
        #pragma unroll
        for (int k = 0; k < NC; ++k) {
            const bool f = (bmask >> k) & 1u;
            float v = f ? kl : 0.f;
            #pragma unroll
            for (int off = 16; off; off >>= 1) v += __shfl_xor(v, off, 32);
            const float c = (float)__popcll(__ballot(f));
            if (lane == k) {
                if (b == 0) { rn0 += v; rc0 += c; }
                else        { rn1 += v; rc1 += c; }
            }
        }
    }

    // ---- deterministic block combine -> one 64-float partial row per block ----
    if (lane < 16) {            // lanes 14,15 carry zeros; covers all 64 slots
        float* base = red + wv * PSTRIDE;
        base[lane]      = rn0;  // num, b=0, class=lane
        base[16 + lane] = rn1;  // num, b=1
        base[32 + lane] = rc0;  // cnt, b=0
        base[48 + lane] = rc1;  // cnt, b=1
    }
    __syncthreads();
    if (tid < PSTRIDE) {
        float s = 0.f;
        #pragma unroll
        for (int q = 0; q < BLK / 32; ++q) s += red[q * PSTRIDE + tid];
        partials[(size_t)blockIdx.x * PSTRIDE + tid] = s;
    }
}

// ---------------- Stage 3: WMMA column-sum of 2048x64 partials + final scalar ----------------
// D = ones(16x4) x B(4x16) + C  => every D element = column sum; invariant to B's K-slot order.
__global__ __launch_bounds__(32) void bkd_finalize(const float* __restrict__ partials,
                                                   float* __restrict__ out)
{
    __shared__ float g[64];
    const int lane = threadIdx.x;          // single wave32, EXEC all ones around WMMA
    const int col  = lane & 15;
    const int half = lane >> 4;
    const v2f ones = {1.0f, 1.0f};

    // 4 independent accumulation chains: {num,cnt} x {cols 0-15, cols 16-31}
    v8f acc0 = {}, acc1 = {}, acc2 = {}, acc3 = {};
    const int c0 = col, c1 = 16 + col, c2 = 32 + col, c3 = 48 + col;
    for (int i = 0; i < G2 / 4; ++i) {
        const int r = 4 * i + 2 * half;    // lane pair {L, L+16} covers block-rows {r..r+3}
        const float* row0 = partials + (size_t)r * PSTRIDE;
        const float* row1 = row0 + PSTRIDE;
        v2f b0; b0.x = row0[c0]; b0.y = row1[c0];
        v2f b1; b1.x = row0[c1]; b1.y = row1[c1];
        v2f b2; b2.x = row0[c2]; b2.y = row1[c2];
        v2f b3; b3.x = row0[c3]; b3.y = row1[c3];
        acc0 = __builtin_amdgcn_wmma_f32_16x16x4_f32(false, ones, false, b0, (short)0, acc0, false, false);
        acc1 = __builtin_amdgcn_wmma_f32_16x16x4_f32(false, ones, false, b1, (short)0, acc1, false, false);
        acc2 = __builtin_amdgcn_wmma_f32_16x16x4_f32(false, ones, false, b2, (short)0, acc2, false, false);
        acc3 = __builtin_amdgcn_wmma_f32_16x16x4_f32(false, ones, false, b3, (short)0, acc3, false, false);
    }
    if (lane < 16) {                       // lanes 16-31 hold redundant copies
        g[col]      = acc0[0];             // num, e = 0..15
        g[16 + col] = acc1[0];             // num, e = 16..31
        g[32 + col] = acc2[0];             // cnt, e = 0..15
        g[48 + col] = acc3[0];             // cnt, e = 16..31
    }
    __syncthreads();
    const float num = g[lane];
    const float cnt = g[32 + lane];
    float contrib = (cnt > 0.f) ? num / (14.f * cnt) : 0.f;
    #pragma unroll
    for (int off = 16; off; off >>= 1) contrib += __shfl_xor(contrib, off, 32);
    if (lane == 0) out[0] = contrib;       // LOSS_WEIGHT == 1
}

extern "C" void kernel_launch(void* const* d_in, const int* in_sizes, int n_in,
                              void* d_out, int out_size, void* d_ws, size_t ws_size,
                              hipStream_t stream)
{
    const float* S  = (const float*)d_in[0];
    const float* T  = (const float*)d_in[1];
    const int*   gt = (const int*)d_in[2];
    float* out = (float*)d_out;

    // workspace: [ u8 labels : NTOT bytes ][ partials : G2*64 floats ]  (~2.2 MB total)
    unsigned char* lab = (unsigned char*)d_ws;
    float* partials = (float*)((char*)d_ws + (size_t)NTOT);

    bkd_compact_labels<<<(NTOT + 255) / 256, 256, 0, stream>>>(gt, lab);
    bkd_main<<<G2, BLK, 0, stream>>>(S, T, lab, partials);
    bkd_finalize<<<1, 32, 0, stream>>>(partials, out);
}